// MacroDGRCL_48911087567224
// MI455X (gfx1250) — compile-verified
//
#include <hip/hip_runtime.h>
#include <hip/hip_bf16.h>

#define N_NODES 8192
#define H_DIM   256
#define TOPK    16
#define NSEC    11
#define NEG_SLOPE 0.2f

typedef __attribute__((ext_vector_type(2))) float v2f;
typedef __attribute__((ext_vector_type(8))) float v8f;

// ---------------------------------------------------------------------------
// Kernel A: b1 = W^T a1, b2 = W^T a2, packed into B matrix (256 x 16),
// columns 0 and 1 hold b1/b2, remaining columns zero.
// ---------------------------------------------------------------------------
__global__ __launch_bounds__(256) void build_b_kernel(
    const float* __restrict__ W, const float* __restrict__ a,
    float* __restrict__ Bm) {
  int k = threadIdx.x;  // 0..255, one output element per thread
  float b1 = 0.f, b2 = 0.f;
  for (int c = 0; c < H_DIM; ++c) {
    float w = W[c * H_DIM + k];  // coalesced across k
    b1 += a[c] * w;
    b2 += a[H_DIM + c] * w;
  }
  Bm[k * 16 + 0] = b1;
  Bm[k * 16 + 1] = b2;
#pragma unroll
  for (int n = 2; n < 16; ++n) Bm[k * 16 + n] = 0.f;
}

// ---------------------------------------------------------------------------
// Kernel B: S = E (8192x256) @ B (256x16) via V_WMMA_F32_16X16X4_F32.
// Each wave32 owns one 16-row tile; 64 WMMA steps over K. Column 0 -> s_i,
// column 1 -> s_j. EXEC is all-ones (no divergence before the WMMAs).
// ---------------------------------------------------------------------------
__global__ __launch_bounds__(256) void dual_matvec_wmma_kernel(
    const float* __restrict__ E, const float* __restrict__ Bm,
    float* __restrict__ si, float* __restrict__ sj) {
  int wave  = threadIdx.x >> 5;
  int lane  = threadIdx.x & 31;
  int m0    = (blockIdx.x * 8 + wave) * 16;   // row tile base
  int m     = lane & 15;                       // row within tile (A) / col (B,D)
  int khalf = lane >> 4;                       // 0: K={0,1}, 1: K={2,3}

  const float* __restrict__ Arow = E + (size_t)(m0 + m) * H_DIM;
  v8f acc = {};
#pragma unroll 4
  for (int k0 = 0; k0 < H_DIM; k0 += 4) {
    int kk = k0 + khalf * 2;
    v2f av, bv;
    av.x = Arow[kk];
    av.y = Arow[kk + 1];
    bv.x = Bm[kk * 16 + m];
    bv.y = Bm[(kk + 1) * 16 + m];
    acc = __builtin_amdgcn_wmma_f32_16x16x4_f32(
        false, av, false, bv, (short)0, acc, false, false);
  }

  // D layout: VGPR r -> M=r (lanes 0-15) / M=r+8 (lanes 16-31), N = lane&15.
  if (m < 2) {
    float* __restrict__ dst = (m == 0) ? si : sj;
    int rbase = m0 + khalf * 8;
#pragma unroll
    for (int r = 0; r < 8; ++r) dst[rbase + r] = acc[r];
  }
}

// ---------------------------------------------------------------------------
// Kernel C: per-sector top-16 of s_j over {j : active[j] && sector[j]==s}.
// Key = (monotone_f32(val) << 32) | (0xFFFFFFFF - j): descending-key order
// == value descending, index ascending (matches jax.lax.top_k tie-breaking).
// Sentinel 0 < any real key. Register top-16 per thread, LDS tree merge.
// Also records the per-sector -inf fill indices (smallest j outside the
// valid set) used when a sector has fewer than 16 valid members.
// ---------------------------------------------------------------------------
__global__ __launch_bounds__(256) void sector_topk_kernel(
    const float* __restrict__ sj, const int* __restrict__ sector,
    const unsigned char* __restrict__ active,
    float* __restrict__ sec_val, int* __restrict__ sec_idx,
    int* __restrict__ sec_cnt, int* __restrict__ sec_fill) {
  int s = blockIdx.x;
  int t = threadIdx.x;

  unsigned long long loc[TOPK];
#pragma unroll
  for (int m = 0; m < TOPK; ++m) loc[m] = 0ull;

  for (int j = t; j < N_NODES; j += 256) {
    if (active[j] && sector[j] == s) {
      unsigned u  = __float_as_uint(sj[j]);
      unsigned mk = (u & 0x80000000u) ? ~u : (u | 0x80000000u);
      unsigned long long key =
          ((unsigned long long)mk << 32) |
          (unsigned long long)(0xFFFFFFFFu - (unsigned)j);
      if (key > loc[TOPK - 1]) {
        int p = TOPK - 1;
        while (p > 0 && loc[p - 1] < key) { loc[p] = loc[p - 1]; --p; }
        loc[p] = key;
      }
    }
  }

  __shared__ unsigned long long sm[256 * TOPK];  // 32 KB of 320 KB LDS
#pragma unroll
  for (int m = 0; m < TOPK; ++m) sm[t * TOPK + m] = loc[m];
  __syncthreads();

  for (int stride = 128; stride >= 1; stride >>= 1) {
    if (t < stride) {
      unsigned long long* A  = &sm[t * TOPK];
      unsigned long long* Bp = &sm[(t + stride) * TOPK];
      unsigned long long out[TOPK];
      int ia = 0, ib = 0;
#pragma unroll
      for (int m = 0; m < TOPK; ++m) {
        unsigned long long av = A[ia], bv = Bp[ib];
        if (av >= bv) { out[m] = av; ++ia; } else { out[m] = bv; ++ib; }
      }
#pragma unroll
      for (int m = 0; m < TOPK; ++m) A[m] = out[m];
    }
    __syncthreads();
  }

  if (t == 0) {
    int cnt = 0;
    for (int m = 0; m < TOPK; ++m) {
      unsigned long long key = sm[m];
      if (key != 0ull) {
        int idx = (int)(0xFFFFFFFFu - (unsigned)(key & 0xFFFFFFFFull));
        sec_idx[s * TOPK + m] = idx;
        sec_val[s * TOPK + m] = sj[idx];
        ++cnt;
      } else {
        sec_idx[s * TOPK + m] = 0;
        sec_val[s * TOPK + m] = 0.f;
      }
    }
    sec_cnt[s] = cnt;
    int nf = 0;
    for (int j = 0; j < N_NODES && nf < TOPK; ++j) {
      if (!(active[j] && sector[j] == s)) sec_fill[s * TOPK + nf++] = j;
    }
    for (; nf < TOPK; ++nf) sec_fill[s * TOPK + nf] = 0;
  }
}

// ---------------------------------------------------------------------------
// Kernel D: emit per-row outputs. Row i just applies leaky_relu to
// (s_i[i] + sector_top_val[m]) for its sector's shared top-16 list.
// Output = 3 flat float segments: weights, float(index), valid(0/1).
// ---------------------------------------------------------------------------
__global__ __launch_bounds__(256) void emit_kernel(
    const float* __restrict__ si, const int* __restrict__ sector,
    const unsigned char* __restrict__ active,
    const float* __restrict__ sec_val, const int* __restrict__ sec_idx,
    const int* __restrict__ sec_cnt, const int* __restrict__ sec_fill,
    float* __restrict__ out) {
  int i = blockIdx.x * 256 + threadIdx.x;  // 32 blocks x 256 = 8192 exact
  float* __restrict__ out_w = out;
  float* __restrict__ out_i = out + (size_t)N_NODES * TOPK;
  float* __restrict__ out_v = out + 2 * (size_t)N_NODES * TOPK;

  bool act   = active[i] != 0;
  int  s     = sector[i];
  float v_si = si[i];
  int  cnt   = act ? sec_cnt[s] : 0;

#pragma unroll
  for (int m = 0; m < TOPK; ++m) {
    float w, fi, fv;
    if (act && m < cnt) {
      float x = v_si + sec_val[s * TOPK + m];
      w  = (x > 0.f) ? x : (NEG_SLOPE * x);   // TEMPERATURE == 1
      fi = (float)sec_idx[s * TOPK + m];
      fv = 1.f;
    } else if (act) {
      w  = 0.f;
      fi = (float)sec_fill[s * TOPK + (m - cnt)];
      fv = 0.f;
    } else {
      w  = 0.f;
      fi = (float)m;   // all-(-inf) row: jax fills indices 0..15
      fv = 0.f;
    }
    size_t o = (size_t)i * TOPK + m;
    out_w[o] = w;
    out_i[o] = fi;
    out_v[o] = fv;
  }
}

// ---------------------------------------------------------------------------
extern "C" void kernel_launch(void* const* d_in, const int* in_sizes, int n_in,
                              void* d_out, int out_size, void* d_ws, size_t ws_size,
                              hipStream_t stream) {
  const float*         E      = (const float*)d_in[0];        // 8192*256
  const float*         W      = (const float*)d_in[1];        // 256*256
  const float*         a      = (const float*)d_in[2];        // 512
  const int*           sector = (const int*)d_in[3];          // 8192
  const unsigned char* active = (const unsigned char*)d_in[4];// 8192 (bool)

  float* ws       = (float*)d_ws;
  float* Bm       = ws;                         // 256*16
  float* s_i      = Bm + 256 * 16;              // 8192
  float* s_j      = s_i + N_NODES;              // 8192
  float* sec_val  = s_j + N_NODES;              // 11*16
  int*   sec_idx  = (int*)(sec_val + NSEC * TOPK);
  int*   sec_cnt  = sec_idx + NSEC * TOPK;      // 11 (padded)
  int*   sec_fill = sec_cnt + 16;               // 11*16

  build_b_kernel<<<1, 256, 0, stream>>>(W, a, Bm);
  dual_matvec_wmma_kernel<<<N_NODES / (16 * 8), 256, 0, stream>>>(E, Bm, s_i, s_j);
  sector_topk_kernel<<<NSEC, 256, 0, stream>>>(s_j, sector, active,
                                               sec_val, sec_idx, sec_cnt, sec_fill);
  emit_kernel<<<N_NODES / 256, 256, 0, stream>>>(s_i, sector, active,
                                                 sec_val, sec_idx, sec_cnt, sec_fill,
                                                 (float*)d_out);
}